// AgroNetLoss_12086037971202
// MI455X (gfx1250) — compile-verified
//
#include <hip/hip_runtime.h>
#include <math.h>

// ---- problem constants (mirror reference) ----
#define NCC      20
#define REGMAX   16
#define AANCH    8400
#define BATCH    32
#define MAXGT    128
#define TOPKK    10
#define GRIDX    33            // ceil(8400/256)
#define NPART    (GRIDX * BATCH)

typedef __attribute__((ext_vector_type(2))) float v2f;
typedef __attribute__((ext_vector_type(8))) float v8f;

static_assert((size_t)BATCH * AANCH * 4 == 1075200, "item count");
static_assert(1075200 % (16 * 8) == 0, "decode tiling");   // 16 items/wave, 8 waves/block

__device__ __forceinline__ float clamp01(float x) { return fminf(fmaxf(x, 0.f), 1.f); }

// align metric shared by thresh + assign kernels (identical codegen => identical bits)
__device__ __forceinline__ float calc_align(float p0, float p1, float p2, float p3,
                                            float g0, float g1, float g2, float g3,
                                            float ax, float ay, float clsv, bool valid,
                                            float* iou_out, bool* inb_out) {
    bool inb = (ax > g0) && (ax < g2) && (ay > g1) && (ay < g3) && valid;
    float ix1 = fmaxf(p0, g0), iy1 = fmaxf(p1, g1);
    float ix2 = fminf(p2, g2), iy2 = fminf(p3, g3);
    float inter = fmaxf(ix2 - ix1, 0.f) * fmaxf(iy2 - iy1, 0.f);
    float a1 = fmaxf((p2 - p0) * (p3 - p1), 0.f);
    float a2 = fmaxf((g2 - g0) * (g3 - g1), 0.f);
    float iou = inter / (a1 + a2 - inter + 1e-7f);
    float cl = clamp01(clsv);
    float iu = clamp01(iou);
    float i2 = iu * iu;
    float al = sqrtf(cl) * (i2 * i2 * i2);
    al = inb ? al : 0.f;
    *iou_out = iou;
    *inb_out = inb;
    return al;
}

// -------- Kernel 1: DFL decode via V_WMMA_F32_16X16X4_F32 chains --------
// per wave: 16 items; E(16x16 exp bins) x [ones | proj] -> s0, s1; d = s1/s0.
__global__ void agro_decode(const float* __restrict__ pdist,
                            const float* __restrict__ apts,
                            const float* __restrict__ strt,
                            float* __restrict__ pred,
                            float* __restrict__ lse) {
    const int lane = threadIdx.x & 31;
    const int wid  = (blockIdx.x * blockDim.x + threadIdx.x) >> 5;
    const int itemBase = wid * 16;
    const int M = lane & 15;
    const int h = lane >> 4;
    const int item = itemBase + M;

    const float* xr = pdist + (size_t)item * REGMAX;
    float ex[8];
    float mx = -3.4e38f;
#pragma unroll
    for (int j = 0; j < 4; ++j) {             // bins 4j+2h, 4j+2h+1
        float x0 = xr[4 * j + 2 * h];
        float x1 = xr[4 * j + 2 * h + 1];
        ex[2 * j] = x0; ex[2 * j + 1] = x1;
        mx = fmaxf(mx, fmaxf(x0, x1));
    }
    mx = fmaxf(mx, __shfl_xor(mx, 16));       // full 16-bin max (lane pair)
    float ssum = 0.f;
#pragma unroll
    for (int k = 0; k < 8; ++k) { ex[k] = expf(ex[k] - mx); ssum += ex[k]; }
    float s0l = ssum + __shfl_xor(ssum, 16);
    float lsev = mx + logf(s0l);
    if (h == 0) lse[item] = lsev;

    // WMMA: A = exp matrix (16x4 f32 per step), B col0 = 1, col1 = proj bin index
    v8f c = {};
#pragma unroll
    for (int j = 0; j < 4; ++j) {
        v2f av; av.x = ex[2 * j]; av.y = ex[2 * j + 1];
        int K0 = 4 * j + 2 * h;
        v2f bv;
        bv.x = (M == 0) ? 1.f : ((M == 1) ? (float)K0 : 0.f);
        bv.y = (M == 0) ? 1.f : ((M == 1) ? (float)(K0 + 1) : 0.f);
        c = __builtin_amdgcn_wmma_f32_16x16x4_f32(false, av, false, bv,
                                                  (short)0, c, false, false);
    }
    // lane0/16 hold s0 of items M=0..7 / 8..15 ; lane1/17 hold s1
    float dv[8];
#pragma unroll
    for (int r = 0; r < 8; ++r) {
        float s1 = __shfl_xor(c[r], 1);
        dv[r] = s1 / c[r];
    }
    if ((lane & 15) == 0) {
        int base = itemBase + (h ? 8 : 0);
#pragma unroll
        for (int r = 0; r < 8; ++r) {
            int it = base + r;
            int c4 = it & 3;
            int a = (it >> 2) % AANCH;
            float sv = strt[a];
            float anc = apts[2 * a + (c4 & 1)];
            float px = (c4 < 2) ? (anc - dv[r]) : (anc + dv[r]);
            pred[it] = px * sv;
        }
    }
}

// -------- Kernel 2: per-(b,g) top-10 threshold over 8400 anchors --------
__global__ void agro_thresh(const float* __restrict__ cls,
                            const float* __restrict__ pred,
                            const float* __restrict__ apts,
                            const float* __restrict__ strt,
                            const float* __restrict__ gtb,
                            const int* __restrict__ gtl,
                            float* __restrict__ thr) {
    __shared__ float als[8448];
    __shared__ float redv[256];
    __shared__ int   redi[256];
    const int tid = threadIdx.x;
    const int bg = blockIdx.x;                       // b*128+g
    const int b = bg >> 7;
    const float g0 = gtb[bg * 4 + 0], g1 = gtb[bg * 4 + 1];
    const float g2 = gtb[bg * 4 + 2], g3 = gtb[bg * 4 + 3];
    const int lr = gtl[bg];
    const bool valid = lr >= 0;
    const int lc = lr < 0 ? 0 : (lr > NCC - 1 ? NCC - 1 : lr);

    for (int it = 0; it < 33; ++it) {
        int a = tid + it * 256;
        float al = 0.f;
        if (a < AANCH) {
            float s = strt[a];
            float ax = apts[2 * a] * s, ay = apts[2 * a + 1] * s;
            const float4 pb = *reinterpret_cast<const float4*>(pred + ((size_t)b * AANCH + a) * 4);
            float clsv = cls[((size_t)b * AANCH + a) * NCC + lc];
            float iou; bool inb;
            al = calc_align(pb.x, pb.y, pb.z, pb.w, g0, g1, g2, g3, ax, ay, clsv, valid, &iou, &inb);
        }
        als[tid + it * 256] = al;
    }
    __syncthreads();
    for (int pass = 0; pass < TOPKK; ++pass) {
        float v = -1.f; int vi = tid;
        for (int it = 0; it < 33; ++it) {
            int idx = tid + it * 256;
            float x = als[idx];
            if (x > v) { v = x; vi = idx; }
        }
        redv[tid] = v; redi[tid] = vi;
        __syncthreads();
        for (int s = 128; s > 0; s >>= 1) {
            if (tid < s && redv[tid + s] > redv[tid]) { redv[tid] = redv[tid + s]; redi[tid] = redi[tid + s]; }
            __syncthreads();
        }
        if (tid == 0) {
            if (pass == TOPKK - 1) thr[bg] = redv[0];
            else als[redi[0]] = -1.f;
        }
        __syncthreads();
    }
}

// -------- Kernel 3: per-anchor assignment with conflict resolution --------
__global__ void agro_assign(const float* __restrict__ cls,
                            const float* __restrict__ pred,
                            const float* __restrict__ apts,
                            const float* __restrict__ strt,
                            const float* __restrict__ gtb,
                            const int* __restrict__ gtl,
                            const float* __restrict__ thr,
                            float* __restrict__ softa,
                            int* __restrict__ fglbl,
                            float* __restrict__ tbox) {
    __shared__ float sg[MAXGT * 4];
    __shared__ int   sl[MAXGT];
    __shared__ float sth[MAXGT];
    const int b = blockIdx.y;
    const int tid = threadIdx.x;
    if (tid < MAXGT) {
        int bg = b * MAXGT + tid;
        sg[tid * 4 + 0] = gtb[bg * 4 + 0];
        sg[tid * 4 + 1] = gtb[bg * 4 + 1];
        sg[tid * 4 + 2] = gtb[bg * 4 + 2];
        sg[tid * 4 + 3] = gtb[bg * 4 + 3];
        sl[tid] = gtl[bg];
        sth[tid] = thr[bg];
    }
    __syncthreads();
    int a = blockIdx.x * 256 + tid;
    if (a >= AANCH) return;
    size_t ia = (size_t)b * AANCH + a;
    float s = strt[a];
    float ax = apts[2 * a] * s, ay = apts[2 * a + 1] * s;
    float4 pb = *reinterpret_cast<const float4*>(pred + ia * 4);
    const float* crow = cls + ia * NCC;

    int n = 0, firstg = 0, bestg = 0;
    float firstiou = 0.f, bestiou = 0.f, bestv = -1.f;
    for (int g = 0; g < MAXGT; ++g) {
        int lr = sl[g];
        bool valid = lr >= 0;
        int lc = lr < 0 ? 0 : (lr > NCC - 1 ? NCC - 1 : lr);
        float iou; bool inb;
        float al = calc_align(pb.x, pb.y, pb.z, pb.w,
                              sg[g * 4 + 0], sg[g * 4 + 1], sg[g * 4 + 2], sg[g * 4 + 3],
                              ax, ay, crow[lc], valid, &iou, &inb);
        bool claim = inb && (al >= sth[g]);
        if (claim) { if (n == 0) { firstg = g; firstiou = iou; } n++; }
        if (al > bestv) { bestv = al; bestg = g; bestiou = iou; }
    }
    bool fg = n > 0;
    int asg = (n > 1) ? bestg : firstg;
    float miou = (n > 1) ? bestiou : firstiou;
    float soft = (bestv / fmaxf(bestv, 1e-9f)) * miou;
    softa[ia] = fg ? soft : 0.f;
    int lr = sl[asg];
    int lab = lr < 0 ? 0 : (lr > NCC - 1 ? NCC - 1 : lr);
    fglbl[ia] = fg ? lab : -1;
    float4 tb;
    if (fg) { tb.x = sg[asg * 4 + 0]; tb.y = sg[asg * 4 + 1]; tb.z = sg[asg * 4 + 2]; tb.w = sg[asg * 4 + 3]; }
    else { tb.x = tb.y = tb.z = tb.w = 0.f; }
    *reinterpret_cast<float4*>(tbox + ia * 4) = tb;
}

// -------- Kernel 4: per-anchor losses + block partial sums --------
__global__ void agro_loss(const float* __restrict__ cls,
                          const float* __restrict__ pdist,
                          const float* __restrict__ apts,
                          const float* __restrict__ strt,
                          const float* __restrict__ pred,
                          const float* __restrict__ lse,
                          const float* __restrict__ softa,
                          const int* __restrict__ fglbl,
                          const float* __restrict__ tbox,
                          float* __restrict__ partial) {
    const int b = blockIdx.y;
    const int tid = threadIdx.x;
    const int a = blockIdx.x * 256 + tid;
    float v_tss = 0.f, v_cls = 0.f, v_box = 0.f, v_dfl = 0.f, v_asp = 0.f, v_gc = 0.f;
    if (a < AANCH) {
        size_t ia = (size_t)b * AANCH + a;
        float4 pb = *reinterpret_cast<const float4*>(pred + ia * 4);
        float4 tb = *reinterpret_cast<const float4*>(tbox + ia * 4);
        float soft = softa[ia];
        int fl = fglbl[ia];
        bool fg = fl >= 0;
        float w = fg ? soft : 0.f;
        v_tss = w;
        // ---- BCE cls ----
        const float* cp = cls + ia * NCC;
        float csum = 0.f;
        for (int c = 0; c < NCC; ++c) {
            float p = fminf(fmaxf(cp[c], 1e-7f), 1.f - 1e-7f);
            float lom = logf(1.f - p);
            if (fg && c == fl) csum += -(soft * logf(p) + (1.f - soft) * lom);
            else csum += -lom;
        }
        v_cls = csum;
        // ---- CIoU ----
        {
            float inter = fmaxf(fminf(pb.z, tb.z) - fmaxf(pb.x, tb.x), 0.f) *
                          fmaxf(fminf(pb.w, tb.w) - fmaxf(pb.y, tb.y), 0.f);
            float pw = fmaxf(pb.z - pb.x, 0.f), ph = fmaxf(pb.w - pb.y, 0.f);
            float tw = fmaxf(tb.z - tb.x, 0.f), th = fmaxf(tb.w - tb.y, 0.f);
            float uni = pw * ph + tw * th - inter + 1e-7f;
            float iou = inter / uni;
            float dx = (pb.x + pb.z) * 0.5f - (tb.x + tb.z) * 0.5f;
            float dy = (pb.y + pb.w) * 0.5f - (tb.y + tb.w) * 0.5f;
            float d2 = dx * dx + dy * dy;
            float ew = fmaxf(fmaxf(pb.z, tb.z) - fminf(pb.x, tb.x), 0.f);
            float eh = fmaxf(fmaxf(pb.w, tb.w) - fminf(pb.y, tb.y), 0.f);
            float c2 = ew * ew + eh * eh + 1e-7f;
            float at = atanf(tw / (th + 1e-7f)) - atanf(pw / (ph + 1e-7f));
            float v = 0.4052847345693511f * at * at;          // 4/pi^2
            float av = v / (1.f - iou + v + 1e-7f);
            v_box = (1.f - (iou - d2 / c2 - av * v)) * w;
        }
        // ---- DFL ----
        {
            float s = strt[a];
            float axp = apts[2 * a] * s, ayp = apts[2 * a + 1] * s;
            float tg[4];
            tg[0] = (axp - tb.x) / s; tg[1] = (ayp - tb.y) / s;
            tg[2] = (tb.z - axp) / s; tg[3] = (tb.w - ayp) / s;
            float dsum = 0.f;
#pragma unroll
            for (int c = 0; c < 4; ++c) {
                float t = fminf(fmaxf(tg[c], 0.f), (float)(REGMAX - 1) - 0.01f);
                int tl = (int)t;
                float wl = (float)(tl + 1) - t;
                const float* xr = pdist + (ia * 4 + c) * REGMAX;
                float lv = lse[ia * 4 + c];
                float cel = lv - xr[tl];
                float cer = lv - xr[tl + 1];
                dsum += cel * wl + cer * (1.f - wl);
            }
            v_dfl = dsum * 0.25f * w;
        }
        // ---- aspect prior ----
        {
            float pw = fmaxf(pb.z - pb.x, 1e-4f), ph = fmaxf(pb.w - pb.y, 1e-4f);
            float gw = fmaxf(tb.z - tb.x, 1e-4f), gh = fmaxf(tb.w - tb.y, 1e-4f);
            bool gate = (gh / gw >= 1.2f) && fg;
            float inter = fmaxf(fminf(pb.z, tb.z) - fmaxf(pb.x, tb.x), 0.f) *
                          fmaxf(fminf(pb.w, tb.w) - fmaxf(pb.y, tb.y), 0.f);
            float a1 = fmaxf((pb.z - pb.x) * (pb.w - pb.y), 0.f);
            float a2 = fmaxf((tb.z - tb.x) * (tb.w - tb.y), 0.f);
            float iouw = inter / (a1 + a2 - inter + 1e-7f);
            float pen = fmaxf(1.5f - ph / pw, 0.f) * (1.f - clamp01(iouw));
            v_asp = gate ? pen : 0.f;
            v_gc = gate ? 1.f : 0.f;
        }
    }
    __shared__ float red[256];
    float vals[6] = {v_tss, v_cls, v_box, v_dfl, v_asp, v_gc};
    int blk = blockIdx.y * gridDim.x + blockIdx.x;
#pragma unroll
    for (int k = 0; k < 6; ++k) {
        red[tid] = vals[k];
        __syncthreads();
        for (int s = 128; s > 0; s >>= 1) {
            if (tid < s) red[tid] += red[tid + s];
            __syncthreads();
        }
        if (tid == 0) partial[k * NPART + blk] = red[0];
        __syncthreads();
    }
}

// -------- Kernel 5: deterministic final reduction --------
__global__ void agro_final(const float* __restrict__ partial, float* __restrict__ out) {
    __shared__ float red[256];
    __shared__ float sums[6];
    const int tid = threadIdx.x;
    float acc[6] = {0.f, 0.f, 0.f, 0.f, 0.f, 0.f};
    for (int i = tid; i < NPART; i += 256)
#pragma unroll
        for (int k = 0; k < 6; ++k) acc[k] += partial[k * NPART + i];
#pragma unroll
    for (int k = 0; k < 6; ++k) {
        red[tid] = acc[k];
        __syncthreads();
        for (int s = 128; s > 0; s >>= 1) {
            if (tid < s) red[tid] += red[tid + s];
            __syncthreads();
        }
        if (tid == 0) sums[k] = red[0];
        __syncthreads();
    }
    if (tid == 0) {
        float tss = fmaxf(sums[0], 1.f);
        float total = 7.5f * sums[2] / tss + 0.5f * sums[1] / tss +
                      1.5f * sums[3] / tss + 0.1f * sums[4] / fmaxf(sums[5], 1.f);
        out[0] = total;
    }
}

extern "C" void kernel_launch(void* const* d_in, const int* in_sizes, int n_in,
                              void* d_out, int out_size, void* d_ws, size_t ws_size,
                              hipStream_t stream) {
    (void)in_sizes; (void)n_in; (void)out_size; (void)ws_size;
    const float* cls   = (const float*)d_in[0];   // [32,8400,20]
    const float* pdist = (const float*)d_in[1];   // [32,8400,4,16]
    const float* apts  = (const float*)d_in[2];   // [8400,2]
    const float* strt  = (const float*)d_in[3];   // [8400,1]
    const float* gtb   = (const float*)d_in[4];   // [32,128,4]
    const int*   gtl   = (const int*)d_in[5];     // [32,128]

    const size_t N_BA   = (size_t)BATCH * AANCH;  // 268800
    const size_t N_ITEM = N_BA * 4;               // 1075200
    float* ws_pred = (float*)d_ws;
    float* ws_lse  = ws_pred + N_ITEM;
    float* ws_thr  = ws_lse + N_ITEM;             // 4096
    float* ws_soft = ws_thr + (size_t)BATCH * MAXGT;
    float* ws_tbox = ws_soft + N_BA;
    float* ws_part = ws_tbox + N_ITEM;            // 6*NPART
    int*   ws_lbl  = (int*)(ws_part + 6 * NPART);

    agro_decode<<<8400, 256, 0, stream>>>(pdist, apts, strt, ws_pred, ws_lse);
    agro_thresh<<<BATCH * MAXGT, 256, 0, stream>>>(cls, ws_pred, apts, strt, gtb, gtl, ws_thr);
    agro_assign<<<dim3(GRIDX, BATCH), 256, 0, stream>>>(cls, ws_pred, apts, strt, gtb, gtl,
                                                        ws_thr, ws_soft, ws_lbl, ws_tbox);
    agro_loss<<<dim3(GRIDX, BATCH), 256, 0, stream>>>(cls, pdist, apts, strt, ws_pred, ws_lse,
                                                      ws_soft, ws_lbl, ws_tbox, ws_part);
    agro_final<<<1, 256, 0, stream>>>(ws_part, (float*)d_out);
}